// MSERankLoss_63316407877851
// MI455X (gfx1250) — compile-verified
//
#include <hip/hip_runtime.h>
#include <hip/hip_bf16.h>

typedef __attribute__((ext_vector_type(2))) float v2f;
typedef __attribute__((ext_vector_type(8))) float v8f;

#define ALPHA_F     3.0f
#define MIN_DIFF_F  0.1f
#define NELEM       8192
#define NT          512          /* 8192 / 16 tiles per dimension */
#define PAIR_BLOCKS 1024
#define PAIR_THREADS 256

#define LOG2E_F 1.4426950408889634f
#define LN2_F   0.6931471805599453f

// Each wave owns 16x16 (i,j) tiles. Pairwise differences are produced as a
// rank-2 outer product with V_WMMA_F32_16X16X4_F32:
//   A(16x4) = [ x_i, 1, 0, 0 ]   B(4x16) = [ 1 ; -x_j ; 0 ; 0 ]
//   D[i][j] = x_i - x_j
// A/B f32 16x4 layout (ISA 7.12.2): lanes 0-15 hold K=0 (VGPR0) and K=1
// (VGPR1); lanes 16-31 hold K=2,3 which we set to zero. D lands in the C/D
// layout: lane = column (N = lane&15), VGPR k = row (M = k, or k+8 for the
// upper half-wave), so each lane owns 8 pairs for the elementwise BPR math.
__global__ __launch_bounds__(PAIR_THREADS) void mserank_pair_tiles(
    const float* __restrict__ pred, const float* __restrict__ tgt,
    float* __restrict__ blk_sum, unsigned* __restrict__ blk_cnt)
{
    const int lane   = threadIdx.x & 31;
    const int wid    = threadIdx.x >> 5;
    const int gwave  = blockIdx.x * (PAIR_THREADS >> 5) + wid;   // wave-uniform
    const int nwaves = gridDim.x * (PAIR_THREADS >> 5);
    const bool lo    = (lane < 16);
    const int  l15   = lane & 15;

    float    lsum = 0.0f;
    unsigned lcnt = 0u;
    const v8f zeroC = {};

    for (int t = gwave; t < NT * NT; t += nwaves) {
        const int tib = t >> 9;          // tile row (i blocks)
        const int tjb = t & (NT - 1);    // tile col (j blocks)
        if (tib > tjb) continue;         // upper triangle only; wave-uniform branch

        const int bi = tib << 4;
        const int bj = tjb << 4;

        // All lanes issue in-range loads (data is L0/L2 resident); lanes 16-31
        // are zeroed by v_cndmask to form the K=2,3 slices.
        const float pi = pred[bi + l15];
        const float pj = pred[bj + l15];
        const float qi = tgt [bi + l15];
        const float qj = tgt [bj + l15];

        v2f aP, bP, aT, bT;
        aP.x = lo ? pi  : 0.0f;  aP.y = lo ? 1.0f : 0.0f;
        bP.x = lo ? 1.0f : 0.0f; bP.y = lo ? -pj  : 0.0f;
        aT.x = lo ? qi  : 0.0f;  aT.y = lo ? 1.0f : 0.0f;
        bT.x = lo ? 1.0f : 0.0f; bT.y = lo ? -qj  : 0.0f;

        // EXEC is all-1s here (only wave-uniform control flow above).
        v8f dP = __builtin_amdgcn_wmma_f32_16x16x4_f32(
            false, aP, false, bP, (short)0, zeroC, false, false);
        v8f dT = __builtin_amdgcn_wmma_f32_16x16x4_f32(
            false, aT, false, bT, (short)0, zeroC, false, false);

        #pragma unroll
        for (int k = 0; k < 8; ++k) {
            const int row = lo ? k : (k + 8);
            const int gi  = bi + row;
            const int gj  = bj + l15;
            const float dt = dT[k];           // target[i] - target[j]
            const float dp = dP[k];           // pred[i]   - pred[j]
            const float c  = fabsf(dt);
            // x = dp * sign(dt): pairs with dt==0 are masked out (c <= MIN_DIFF),
            // so copysign (one v_bfi_b32) is equivalent under the mask and avoids
            // the cmp/cndmask sign-select chain.
            const float x  = copysignf(dp, dt);
            // -log_sigmoid(x) = softplus(-x) = max(-x,0) + ln(1 + exp(-|x|)),
            // with |x| == |dp|. Use the raw hardware trans ops: the exp2 input is
            // <= 0 (no overflow path) and the log input is in (1,2] (no denorm
            // path), so the library guard code around v_exp_f32/v_log_f32 is
            // unnecessary.
            const float e  = __builtin_amdgcn_exp2f(-fabsf(dp) * LOG2E_F); // v_exp_f32
            const float lg = __builtin_amdgcn_logf(1.0f + e);              // v_log_f32 (log2)
            const float sp = fmaf(LN2_F, lg, fmaxf(-x, 0.0f));
            if ((gi < gj) && (c > MIN_DIFF_F)) {
                lsum = fmaf(c, sp, lsum);
                lcnt++;
            }
        }
    }

    // wave32 butterfly reduce
    #pragma unroll
    for (int off = 16; off > 0; off >>= 1) {
        lsum += __shfl_down(lsum, off, 32);
        lcnt += __shfl_down(lcnt, off, 32);
    }

    __shared__ float    ssum[PAIR_THREADS / 32];
    __shared__ unsigned scnt[PAIR_THREADS / 32];
    if (lane == 0) { ssum[wid] = lsum; scnt[wid] = lcnt; }
    __syncthreads();
    if (threadIdx.x == 0) {
        float    bs = 0.0f;
        unsigned bc = 0u;
        #pragma unroll
        for (int w = 0; w < PAIR_THREADS / 32; ++w) { bs += ssum[w]; bc += scnt[w]; }
        blk_sum[blockIdx.x] = bs;   // fully rewritten every call: no stale ws state
        blk_cnt[blockIdx.x] = bc;
    }
}

// One block: MSE term over 8192 elems + deterministic combine of the 1024
// block partials (double accumulation for the 33.5M-term pair sum).
__global__ __launch_bounds__(256) void mserank_finalize(
    const float* __restrict__ pred, const float* __restrict__ tgt,
    const float* __restrict__ blk_sum, const unsigned* __restrict__ blk_cnt,
    float* __restrict__ out)
{
    const int tid = threadIdx.x;

    float mse = 0.0f;
    for (int i = tid; i < NELEM; i += 256) {
        const float d = pred[i] - tgt[i];
        mse = fmaf(d, d, mse);
    }

    double             ps = 0.0;
    unsigned long long pc = 0ull;
    for (int b = tid; b < PAIR_BLOCKS; b += 256) {
        ps += (double)blk_sum[b];
        pc += (unsigned long long)blk_cnt[b];
    }

    __shared__ float              smse[256];
    __shared__ double             sps [256];
    __shared__ unsigned long long spc [256];
    smse[tid] = mse; sps[tid] = ps; spc[tid] = pc;
    __syncthreads();
    for (int off = 128; off > 0; off >>= 1) {
        if (tid < off) {
            smse[tid] += smse[tid + off];
            sps [tid] += sps [tid + off];
            spc [tid] += spc [tid + off];
        }
        __syncthreads();
    }

    if (tid == 0) {
        const float reg = smse[0] / (float)NELEM;
        float res = reg;
        if (spc[0] > 0ull) {
            res = reg + ALPHA_F * (float)(sps[0] / (double)spc[0]);
        }
        out[0] = res;
    }
}

extern "C" void kernel_launch(void* const* d_in, const int* in_sizes, int n_in,
                              void* d_out, int out_size, void* d_ws, size_t ws_size,
                              hipStream_t stream) {
    (void)in_sizes; (void)n_in; (void)out_size; (void)ws_size;
    const float* pred = (const float*)d_in[0];
    const float* tgt  = (const float*)d_in[1];
    float*    out     = (float*)d_out;
    float*    blk_sum = (float*)d_ws;                       // 1024 floats
    unsigned* blk_cnt = (unsigned*)(blk_sum + PAIR_BLOCKS); // 1024 uints (8 KB total)

    mserank_pair_tiles<<<PAIR_BLOCKS, PAIR_THREADS, 0, stream>>>(pred, tgt, blk_sum, blk_cnt);
    mserank_finalize<<<1, 256, 0, stream>>>(pred, tgt, blk_sum, blk_cnt, out);
}